// PoseEncoder_87995289960562
// MI455X (gfx1250) — compile-verified
//
#include <hip/hip_runtime.h>
#include <cstdint>
#include <cstddef>

// ---------------------------------------------------------------------------
// PoseEncoder: out[b, k*3+i] = (inv(fwd[b,k]) @ [trans[b],1])[i], i<3
// Memory-bound: ~480MB traffic -> ~21us at 23.3TB/s. Compute via adjugate
// (Cramer) solve, ~120 flops/matrix -> far below fp32 VALU ceiling.
// Data movement: TDM (tensor_load_to_lds) stages contiguous 32KB fwd tiles
// into LDS with 16B padding per 256B to kill ds_load_b128 bank conflicts;
// LDS is read back with explicit inline-asm ds_load_b128 (the TDM
// intrinsic's LDS writes are invisible to LLVM alias/escape analysis, so
// ordinary C++ loads of smem fold to poison and everything gets DCE'd --
// observed rounds 1-2). Outputs use dense nontemporal b128 stores.
// ---------------------------------------------------------------------------

typedef __attribute__((ext_vector_type(4))) float        f32x4;
typedef __attribute__((ext_vector_type(4))) unsigned int u32x4;
typedef __attribute__((ext_vector_type(4))) int          i32x4;
typedef __attribute__((ext_vector_type(8))) int          i32x8;

#define THREADS          128
#define MATS_PER_THREAD  4
#define TILE_BYTES       (THREADS * MATS_PER_THREAD * 64)   // 32768 B per block
#define LDS_CHUNK_B      272                                // 256 data + 16 pad
#define LDS_V4           (THREADS * LDS_CHUNK_B / 16)       // 2176 f32x4

#define TOTAL_MATS       (262144 * 24)                      // B*K = 6291456
#define TENSOR_QWORDS    (TOTAL_MATS * 8u)                  // 50331648 8B elems

__global__ __launch_bounds__(THREADS)
void pose_encoder_kernel(const float* __restrict__ trans,
                         const float* __restrict__ fwd,
                         float* __restrict__ out)
{
    __shared__ f32x4 smem[LDS_V4];

    const uint32_t tid = threadIdx.x;

    // ---- TDM: DMA one contiguous 32KB tile of fwd into LDS (wave 0 only) ----
    if (tid == 0) {
        const uint64_t ga = (uint64_t)(uintptr_t)fwd +
                            (uint64_t)blockIdx.x * (uint64_t)TILE_BYTES;
        // D# group 0: count=1, lds_addr=0, global_addr, type=2
        u32x4 g0;
        g0.x = 1u;                                   // count=1 (valid descriptor)
        g0.y = 0u;                                   // lds_addr = 0 (sole LDS alloc)
        g0.z = (uint32_t)ga;                         // global_addr[31:0]
        g0.w = (uint32_t)((ga >> 32) & 0x1FFFFFFu)   // global_addr[56:32]
               | 0x80000000u;                        // type=2 ("image")
        // D# group 1: data_size=8B, pad 4 dwords every 64 dwords (16B/256B),
        // 1-D tensor of TENSOR_QWORDS elems, tile_dim0 = 4096 elems (32KB)
        i32x8 g1;
        g1[0] = (int)((3u << 16)        // data_size = 8B
                      | (1u << 20)      // pad_enable
                      | (5u << 22)      // pad_interval: 64 dwords = 256B
                      | (3u << 25));    // pad_amount:   4 dwords = 16B
        g1[1] = (int)((TENSOR_QWORDS & 0xFFFFu) << 16);          // dim0[15:0]
        g1[2] = (int)(((TENSOR_QWORDS >> 16) & 0xFFFFu)          // dim0[31:16]
                      | (1u << 16));                             // tensor_dim1=1
        g1[3] = (int)(((uint32_t)(TILE_BYTES / 8)) << 16);       // tile_dim0=4096
        g1[4] = (int)(1u | (1u << 16));                          // tile_dim1/2 = 1
        g1[5] = (int)TENSOR_QWORDS;                              // dim0_stride lo
        g1[6] = 0;                                               // stride hi/lo16
        g1[7] = (int)(TENSOR_QWORDS >> 16);                      // dim1_stride
        // D# groups 2/3: higher dims = 1, strides = 0
        i32x4 g2 = { 1, 1, 0, (int)(1u << 16) };
        i32x4 g3 = { 0, (int)(1u << 16), (int)(1u << 16), 0 };
#if defined(__clang_major__) && (__clang_major__ >= 23)
        i32x8 g4 = { 0, 0, 0, 0, 0, 0, 0, 0 };
        __builtin_amdgcn_tensor_load_to_lds(g0, g1, g2, g3, g4, 0);
#else
        __builtin_amdgcn_tensor_load_to_lds(g0, g1, g2, g3, 0);
#endif
        __builtin_amdgcn_s_wait_tensorcnt(0);
    }
    __syncthreads();

    // ---- per-thread: 4 matrices, adjugate solve ----
    const uint32_t g = blockIdx.x * THREADS + tid;   // 4-matrix group id
    const uint32_t b = g / 6;                        // 6 groups of 4 per b (K=24)
    const float tx = trans[b * 3 + 0];
    const float ty = trans[b * 3 + 1];
    const float tz = trans[b * 3 + 2];

    // LDS byte offset of this thread's 256B data chunk (+16B pad per chunk).
    // Truncating the generic pointer yields the LDS offset (ISA 10.2: flat->
    // LDS maps addr[31:0]); feeding it to asm also escapes smem so the
    // allocation and the TDM's writes can't be optimized away.
    const uint32_t laddr = (uint32_t)(uintptr_t)smem + tid * LDS_CHUNK_B;

    float res[12];
#pragma unroll
    for (int j = 0; j < MATS_PER_THREAD; ++j) {
        const uint32_t a = laddr + (uint32_t)(j * 64);
        f32x4 r0, r1, r2, r3;
        asm volatile(
            "ds_load_b128 %0, %4\n\t"
            "ds_load_b128 %1, %4 offset:16\n\t"
            "ds_load_b128 %2, %4 offset:32\n\t"
            "ds_load_b128 %3, %4 offset:48\n\t"
            "s_wait_dscnt 0x0"
            : "=v"(r0), "=v"(r1), "=v"(r2), "=v"(r3)
            : "v"(a));
        // 2x2 minors of rows 0-1 (s) and rows 2-3 (c)
        const float s0 = r0.x * r1.y - r1.x * r0.y;
        const float s1 = r0.x * r1.z - r1.x * r0.z;
        const float s2 = r0.x * r1.w - r1.x * r0.w;
        const float s3 = r0.y * r1.z - r1.y * r0.z;
        const float s4 = r0.y * r1.w - r1.y * r0.w;
        const float s5 = r0.z * r1.w - r1.z * r0.w;
        const float c5 = r2.z * r3.w - r3.z * r2.w;
        const float c4 = r2.y * r3.w - r3.y * r2.w;
        const float c3 = r2.y * r3.z - r3.y * r2.z;
        const float c2 = r2.x * r3.w - r3.x * r2.w;
        const float c1 = r2.x * r3.z - r3.x * r2.z;
        const float c0 = r2.x * r3.y - r3.x * r2.y;
        const float det = s0 * c5 - s1 * c4 + s2 * c3
                        + s3 * c2 - s4 * c1 + s5 * c0;
        const float invdet = 1.0f / det;
        // adjugate rows 0..2 (row 3 of inverse never needed)
        const float i00 =  r1.y * c5 - r1.z * c4 + r1.w * c3;
        const float i01 = -r0.y * c5 + r0.z * c4 - r0.w * c3;
        const float i02 =  r3.y * s5 - r3.z * s4 + r3.w * s3;
        const float i03 = -r2.y * s5 + r2.z * s4 - r2.w * s3;
        const float i10 = -r1.x * c5 + r1.z * c2 - r1.w * c1;
        const float i11 =  r0.x * c5 - r0.z * c2 + r0.w * c1;
        const float i12 = -r3.x * s5 + r3.z * s2 - r3.w * s1;
        const float i13 =  r2.x * s5 - r2.z * s2 + r2.w * s1;
        const float i20 =  r1.x * c4 - r1.y * c2 + r1.w * c0;
        const float i21 = -r0.x * c4 + r0.y * c2 - r0.w * c0;
        const float i22 =  r3.x * s4 - r3.y * s2 + r3.w * s0;
        const float i23 = -r2.x * s4 + r2.y * s2 - r2.w * s0;
        res[j * 3 + 0] = (i00 * tx + i01 * ty + i02 * tz + i03) * invdet;
        res[j * 3 + 1] = (i10 * tx + i11 * ty + i12 * tz + i13) * invdet;
        res[j * 3 + 2] = (i20 * tx + i21 * ty + i22 * tz + i23) * invdet;
    }

    // 12 contiguous floats at float offset 12*g; 48g bytes is 16B aligned ->
    // three dense nontemporal b128 stores (output never re-read).
    f32x4 o0 = { res[0], res[1], res[2],  res[3]  };
    f32x4 o1 = { res[4], res[5], res[6],  res[7]  };
    f32x4 o2 = { res[8], res[9], res[10], res[11] };
    f32x4* op = reinterpret_cast<f32x4*>(out + (size_t)g * 12);
    __builtin_nontemporal_store(o0, op + 0);
    __builtin_nontemporal_store(o1, op + 1);
    __builtin_nontemporal_store(o2, op + 2);
}

extern "C" void kernel_launch(void* const* d_in, const int* in_sizes, int n_in,
                              void* d_out, int out_size, void* d_ws, size_t ws_size,
                              hipStream_t stream) {
    (void)in_sizes; (void)n_in; (void)d_ws; (void)ws_size; (void)out_size;
    const float* trans = (const float*)d_in[0];            // [B,3]
    const float* fwd   = (const float*)d_in[1];            // [B,K,4,4]
    float*       out   = (float*)d_out;                    // [B,K*3]
    const int grid = TOTAL_MATS / (THREADS * MATS_PER_THREAD);  // 12288
    pose_encoder_kernel<<<grid, THREADS, 0, stream>>>(trans, fwd, out);
}